// GraphAttentionLayer_8650064134298
// MI455X (gfx1250) — compile-verified
//
#include <hip/hip_runtime.h>
#include <math.h>

// ---------------- problem constants ----------------
#define NPTS   32768
#define KNN_K  16
#define CIN    64
#define MID    64
#define HCH    8                      // MID / SHARE
#define ROWS   (NPTS * KNN_K)         // 524288
#define XS_STRIDE 68                  // 64 + 4 pad floats (bank-conflict-free frag reads)

typedef __attribute__((ext_vector_type(2))) float v2f;
typedef __attribute__((ext_vector_type(4))) float v4f;
typedef __attribute__((ext_vector_type(8))) float v8f;

// f32 WMMA: D(16x16) = A(16x4) * B(4x16) + C   (emits v_wmma_f32_16x16x4_f32)
__device__ __forceinline__ v8f wmma4(v2f a, v2f b, v8f c) {
  return __builtin_amdgcn_wmma_f32_16x16x4_f32(false, a, false, b, (short)0, c,
                                               false, false);
}

// =====================================================================
// Kernel 1: exact KNN (self included).  1 thread = 1 query, LDS point tiles,
// register insertion-sort top-16.  Neighbor ORDER is irrelevant downstream
// (softmax/sum over K are permutation invariant).
// =====================================================================
__global__ __launch_bounds__(256) void knn_kernel(const float* __restrict__ p,
                                                  int* __restrict__ idx_out,
                                                  float* __restrict__ dist_out) {
  __shared__ float Ps[256 * 3];
  const int t = threadIdx.x;
  const int q = blockIdx.x * 256 + t;
  const float qx = p[q * 3 + 0], qy = p[q * 3 + 1], qz = p[q * 3 + 2];

  float db[KNN_K];
  int   ib[KNN_K];
#pragma unroll
  for (int j = 0; j < KNN_K; ++j) { db[j] = 3.4e38f; ib[j] = 0; }

  for (int base = 0; base < NPTS; base += 256) {
    Ps[t * 3 + 0] = p[(base + t) * 3 + 0];
    Ps[t * 3 + 1] = p[(base + t) * 3 + 1];
    Ps[t * 3 + 2] = p[(base + t) * 3 + 2];
    __syncthreads();
#pragma unroll 4
    for (int j = 0; j < 256; ++j) {
      const float dx = Ps[3 * j + 0] - qx;
      const float dy = Ps[3 * j + 1] - qy;
      const float dz = Ps[3 * j + 2] - qz;
      const float d2 = dx * dx + dy * dy + dz * dz;
      const int cand = base + j;
      if (d2 < db[KNN_K - 1]) {
        const float oldd0 = db[0];
#pragma unroll
        for (int s = KNN_K - 1; s >= 1; --s) {
          const float om = db[s - 1]; const int oim = ib[s - 1];
          const float os = db[s];     const int ois = ib[s];
          const bool gtm = om > d2;
          const bool gts = os > d2;
          db[s] = gtm ? om  : (gts ? d2   : os);
          ib[s] = gtm ? oim : (gts ? cand : ois);
        }
        if (oldd0 > d2) { db[0] = d2; ib[0] = cand; }
      }
    }
    __syncthreads();
  }
#pragma unroll
  for (int j = 0; j < KNN_K; ++j) {
    idx_out[q * KNN_K + j]  = ib[j];
    dist_out[q * KNN_K + j] = sqrtf(fmaxf(db[j], 1e-12f));
  }
}

// =====================================================================
// Kernel 2: x_q = x @ w_q + b_q  via f32 WMMA.  1 wave = one 16x16 tile.
// =====================================================================
__global__ __launch_bounds__(128) void xq_gemm_kernel(const float* __restrict__ x,
                                                      const float* __restrict__ w,
                                                      const float* __restrict__ b,
                                                      float* __restrict__ y) {
  const int wave = threadIdx.x >> 5;
  const int lane = threadIdx.x & 31;
  const int ll = lane & 15, lh = lane >> 4;
  const int tile = blockIdx.x * 4 + wave;
  const int rt = tile >> 2, ct = tile & 3;
  const int col = ct * 16 + ll;

  v8f c;
  const float bc = b[col];
#pragma unroll
  for (int r = 0; r < 8; ++r) c[r] = bc;

#pragma unroll
  for (int kk = 0; kk < 16; ++kk) {
    const int k0 = kk * 4 + lh * 2;
    v2f a, bb;
    const float* ap = x + (size_t)(rt * 16 + ll) * 64 + k0;
    a[0] = ap[0]; a[1] = ap[1];
    bb[0] = w[(k0 + 0) * 64 + col];
    bb[1] = w[(k0 + 1) * 64 + col];
    c = wmma4(a, bb, c);
  }
#pragma unroll
  for (int r = 0; r < 8; ++r)
    y[(size_t)(rt * 16 + r + 8 * lh) * 64 + col] = c[r];
}

// =====================================================================
// Kernel 3: stats for bn_p over t_p = (p[idx]-p) @ w_p0 + b_p0   (3 channels)
// =====================================================================
__global__ __launch_bounds__(256) void stats_p_kernel(const float* __restrict__ p,
                                                      const int* __restrict__ idx,
                                                      const float* __restrict__ w_p0,
                                                      const float* __restrict__ b_p0,
                                                      float* __restrict__ stats) {
  __shared__ float acc[6];
  const int t = threadIdx.x;
  if (t < 6) acc[t] = 0.f;
  __syncthreads();

  float s1[3] = {0, 0, 0}, s2[3] = {0, 0, 0};
  for (int row = blockIdx.x * 256 + t; row < ROWS; row += gridDim.x * 256) {
    const int n = row >> 4;
    const int j = idx[row];
    const float prx = p[j * 3 + 0] - p[n * 3 + 0];
    const float pry = p[j * 3 + 1] - p[n * 3 + 1];
    const float prz = p[j * 3 + 2] - p[n * 3 + 2];
    const float t0 = prx * w_p0[0] + pry * w_p0[3] + prz * w_p0[6] + b_p0[0];
    const float t1 = prx * w_p0[1] + pry * w_p0[4] + prz * w_p0[7] + b_p0[1];
    const float t2 = prx * w_p0[2] + pry * w_p0[5] + prz * w_p0[8] + b_p0[2];
    s1[0] += t0; s2[0] += t0 * t0;
    s1[1] += t1; s2[1] += t1 * t1;
    s1[2] += t2; s2[2] += t2 * t2;
  }
#pragma unroll
  for (int c2 = 0; c2 < 3; ++c2) {
    atomicAdd(&acc[c2], s1[c2]);
    atomicAdd(&acc[3 + c2], s2[c2]);
  }
  __syncthreads();
  if (t < 6) atomicAdd(&stats[t], acc[t]);
}

// =====================================================================
// Generic BN finalize: scale = g*rsqrt(var+eps), shift = b - mean*scale
// =====================================================================
__global__ void finalize_bn_kernel(const float* __restrict__ sum,
                                   const float* __restrict__ sq,
                                   const float* __restrict__ g,
                                   const float* __restrict__ b,
                                   float* __restrict__ sc, float* __restrict__ sh,
                                   int C, float invCount) {
  const int c = threadIdx.x;
  if (c < C) {
    const float mean = sum[c] * invCount;
    const float var  = sq[c] * invCount - mean * mean;
    const float s    = rsqrtf(var + 1e-5f) * g[c];
    sc[c] = s;
    sh[c] = b[c] - mean * s;
  }
}

// ---------------------------------------------------------------------
// Shared phase-A helper: gather x rows + relu(bn_p(t_p)) per row into LDS.
// Block covers 16 queries x 16 neighbors = 256 rows.
// ---------------------------------------------------------------------
__device__ __forceinline__ void phaseA_gather(const float* __restrict__ p,
                                              const float* __restrict__ x,
                                              const int* __restrict__ idx,
                                              const float* __restrict__ w_p0,
                                              const float* __restrict__ b_p0,
                                              const float* __restrict__ scP,
                                              const float* __restrict__ shP,
                                              int q0, int t, float* Xs, float* Q3) {
  const int n = q0 + (t >> 4);
  const int j = idx[n * 16 + (t & 15)];
  const v4f* xr = (const v4f*)(x + (size_t)j * 64);
  v4f* xd = (v4f*)(Xs + t * XS_STRIDE);
#pragma unroll
  for (int i = 0; i < 16; ++i) xd[i] = xr[i];
  const float prx = p[j * 3 + 0] - p[n * 3 + 0];
  const float pry = p[j * 3 + 1] - p[n * 3 + 1];
  const float prz = p[j * 3 + 2] - p[n * 3 + 2];
  const float t0 = prx * w_p0[0] + pry * w_p0[3] + prz * w_p0[6] + b_p0[0];
  const float t1 = prx * w_p0[1] + pry * w_p0[4] + prz * w_p0[7] + b_p0[1];
  const float t2 = prx * w_p0[2] + pry * w_p0[5] + prz * w_p0[8] + b_p0[2];
  Q3[t * 4 + 0] = fmaxf(0.f, t0 * scP[0] + shP[0]);
  Q3[t * 4 + 1] = fmaxf(0.f, t1 * scP[1] + shP[1]);
  Q3[t * 4 + 2] = fmaxf(0.f, t2 * scP[2] + shP[2]);
}

// =====================================================================
// Kernel 4: stats for bn_w0 over w1_pre = x_q - x_k + p_r64 (64 channels).
// x_k via WMMA with NEGATED w_k; C-init = x_q + p_r64 - b_k.
// =====================================================================
__global__ __launch_bounds__(256) void stats_w0_kernel(
    const float* __restrict__ p, const float* __restrict__ x,
    const int* __restrict__ idx, const float* __restrict__ xq,
    const float* __restrict__ w_p0, const float* __restrict__ b_p0,
    const float* __restrict__ w_p1, const float* __restrict__ b_p1,
    const float* __restrict__ w_k, const float* __restrict__ b_k,
    const float* __restrict__ scP, const float* __restrict__ shP,
    float* __restrict__ statsW0) {
  extern __shared__ float sm[];
  float* Xs   = sm;                       // 256*68
  float* Q3   = Xs + 256 * XS_STRIDE;     // 256*4
  float* chan = Q3 + 256 * 4;             // 128
  const int t = threadIdx.x;
  const int q0 = blockIdx.x * 16;
  if (t < 128) chan[t] = 0.f;
  phaseA_gather(p, x, idx, w_p0, b_p0, scP, shP, q0, t, Xs, Q3);
  __syncthreads();

  const int wave = t >> 5, lane = t & 31, ll = lane & 15, lh = lane >> 4;
  for (int tt = wave; tt < 64; tt += 8) {
    const int rt = tt >> 2, ct = tt & 3;
    const int col = ct * 16 + ll;
    const float wc0 = w_p1[col], wc1 = w_p1[64 + col], wc2 = w_p1[128 + col];
    const float cb = xq[(size_t)(q0 + rt) * 64 + col] + b_p1[col] - b_k[col];
    v8f c;
#pragma unroll
    for (int r = 0; r < 8; ++r) {
      const int row = rt * 16 + r + 8 * lh;
      c[r] = cb + Q3[row * 4 + 0] * wc0 + Q3[row * 4 + 1] * wc1 + Q3[row * 4 + 2] * wc2;
    }
#pragma unroll
    for (int kk = 0; kk < 16; ++kk) {
      const int k0 = kk * 4 + lh * 2;
      v2f a, bb;
      a[0] = Xs[(rt * 16 + ll) * XS_STRIDE + k0];
      a[1] = Xs[(rt * 16 + ll) * XS_STRIDE + k0 + 1];
      bb[0] = -w_k[(k0 + 0) * 64 + col];
      bb[1] = -w_k[(k0 + 1) * 64 + col];
      c = wmma4(a, bb, c);
    }
    float s1 = 0.f, s2 = 0.f;
#pragma unroll
    for (int r = 0; r < 8; ++r) { s1 += c[r]; s2 += c[r] * c[r]; }
    atomicAdd(&chan[col], s1);
    atomicAdd(&chan[64 + col], s2);
  }
  __syncthreads();
  if (t < 128) atomicAdd(&statsW0[t], chan[t]);
}

// =====================================================================
// Kernel 5: recompute w1_pre, apply bn_w0+ReLU, compute h = w1a @ w_w0 + b_w0,
// store h [N*K,8] and accumulate bn_w1 stats (8 channels).
// =====================================================================
__global__ __launch_bounds__(256) void stats_w1_kernel(
    const float* __restrict__ p, const float* __restrict__ x,
    const int* __restrict__ idx, const float* __restrict__ xq,
    const float* __restrict__ w_p0, const float* __restrict__ b_p0,
    const float* __restrict__ w_p1, const float* __restrict__ b_p1,
    const float* __restrict__ w_k, const float* __restrict__ b_k,
    const float* __restrict__ scP, const float* __restrict__ shP,
    const float* __restrict__ scW0, const float* __restrict__ shW0,
    const float* __restrict__ w_w0, const float* __restrict__ b_w0,
    float* __restrict__ hout, float* __restrict__ statsW1) {
  extern __shared__ float sm[];
  float* Xs   = sm;                        // 256*68
  float* Q3   = Xs + 256 * XS_STRIDE;      // 256*4
  float* W1s  = Q3 + 256 * 4;              // 256*68
  float* chan = W1s + 256 * XS_STRIDE;     // 16
  const int t = threadIdx.x;
  const int q0 = blockIdx.x * 16;
  if (t < 16) chan[t] = 0.f;
  phaseA_gather(p, x, idx, w_p0, b_p0, scP, shP, q0, t, Xs, Q3);
  __syncthreads();

  const int wave = t >> 5, lane = t & 31, ll = lane & 15, lh = lane >> 4;
  for (int tt = wave; tt < 64; tt += 8) {
    const int rt = tt >> 2, ct = tt & 3;
    const int col = ct * 16 + ll;
    const float wc0 = w_p1[col], wc1 = w_p1[64 + col], wc2 = w_p1[128 + col];
    const float cb = xq[(size_t)(q0 + rt) * 64 + col] + b_p1[col] - b_k[col];
    const float sc = scW0[col], sh = shW0[col];
    v8f c;
#pragma unroll
    for (int r = 0; r < 8; ++r) {
      const int row = rt * 16 + r + 8 * lh;
      c[r] = cb + Q3[row * 4 + 0] * wc0 + Q3[row * 4 + 1] * wc1 + Q3[row * 4 + 2] * wc2;
    }
#pragma unroll
    for (int kk = 0; kk < 16; ++kk) {
      const int k0 = kk * 4 + lh * 2;
      v2f a, bb;
      a[0] = Xs[(rt * 16 + ll) * XS_STRIDE + k0];
      a[1] = Xs[(rt * 16 + ll) * XS_STRIDE + k0 + 1];
      bb[0] = -w_k[(k0 + 0) * 64 + col];
      bb[1] = -w_k[(k0 + 1) * 64 + col];
      c = wmma4(a, bb, c);
    }
#pragma unroll
    for (int r = 0; r < 8; ++r) {
      const int row = rt * 16 + r + 8 * lh;
      W1s[row * XS_STRIDE + col] = fmaxf(0.f, c[r] * sc + sh);
    }
  }
  __syncthreads();

  // per-row: h = w1a @ w_w0 + b_w0   (64 -> 8)
  float h[8];
#pragma unroll
  for (int j = 0; j < 8; ++j) h[j] = b_w0[j];
  for (int cc = 0; cc < 64; ++cc) {
    const float xv = W1s[t * XS_STRIDE + cc];
#pragma unroll
    for (int j = 0; j < 8; ++j) h[j] += xv * w_w0[cc * 8 + j];
  }
  const size_t grow = (size_t)blockIdx.x * 256 + t;
#pragma unroll
  for (int j = 0; j < 8; ++j) {
    hout[grow * 8 + j] = h[j];
    atomicAdd(&chan[j], h[j]);
    atomicAdd(&chan[8 + j], h[j] * h[j]);
  }
  __syncthreads();
  if (t < 16) atomicAdd(&statsW1[t], chan[t]);
}

// =====================================================================
// Kernel 6: final.  softmax(relu(bn_w1(h)) @ w_w1 + b_w1) over K, x_v via WMMA,
// out = sum_k (x_v*exp(-d) + p_r64) * sm
// =====================================================================
__global__ __launch_bounds__(256) void final_kernel(
    const float* __restrict__ p, const float* __restrict__ x,
    const int* __restrict__ idx, const float* __restrict__ dist,
    const float* __restrict__ hbuf,
    const float* __restrict__ w_p0, const float* __restrict__ b_p0,
    const float* __restrict__ w_p1, const float* __restrict__ b_p1,
    const float* __restrict__ w_v, const float* __restrict__ b_v,
    const float* __restrict__ w_w1, const float* __restrict__ b_w1,
    const float* __restrict__ scP, const float* __restrict__ shP,
    const float* __restrict__ scW1, const float* __restrict__ shW1,
    float* __restrict__ out) {
  extern __shared__ float sm[];
  float* Xs  = sm;                        // 256*68
  float* Q3  = Xs + 256 * XS_STRIDE;      // 256*4
  float* Dw  = Q3 + 256 * 4;              // 256
  float* SMs = Dw + 256;                  // 256*8
  float* SMn = SMs + 256 * 8;             // 256*8
  const int t = threadIdx.x;
  const int q0 = blockIdx.x * 16;
  phaseA_gather(p, x, idx, w_p0, b_p0, scP, shP, q0, t, Xs, Q3);
  Dw[t] = expf(-dist[(size_t)blockIdx.x * 256 + t]);
  __syncthreads();

  // attention logits per row
  {
    const size_t grow = (size_t)blockIdx.x * 256 + t;
    float a[8];
#pragma unroll
    for (int j = 0; j < 8; ++j)
      a[j] = fmaxf(0.f, hbuf[grow * 8 + j] * scW1[j] + shW1[j]);
#pragma unroll
    for (int jo = 0; jo < 8; ++jo) {
      float v = b_w1[jo];
#pragma unroll
      for (int j = 0; j < 8; ++j) v += a[j] * w_w1[j * 8 + jo];
      SMs[t * 8 + jo] = v;
    }
  }
  __syncthreads();
  // softmax over the 16 neighbors of this thread's query, per channel
  {
    const int base = (t >> 4) * 16;
#pragma unroll
    for (int jo = 0; jo < 8; ++jo) {
      float m = -3.4e38f;
      for (int i = 0; i < 16; ++i) m = fmaxf(m, SMs[(base + i) * 8 + jo]);
      float s = 0.f;
      for (int i = 0; i < 16; ++i) s += expf(SMs[(base + i) * 8 + jo] - m);
      SMn[t * 8 + jo] = expf(SMs[t * 8 + jo] - m) / s;
    }
  }
  __syncthreads();

  const int wave = t >> 5, lane = t & 31, ll = lane & 15, lh = lane >> 4;
  for (int tt = wave; tt < 64; tt += 8) {
    const int rt = tt >> 2, ct = tt & 3;
    const int col = ct * 16 + ll;
    const float wc0 = w_p1[col], wc1 = w_p1[64 + col], wc2 = w_p1[128 + col];
    const float bp = b_p1[col];
    v8f c;
    const float bv = b_v[col];
#pragma unroll
    for (int r = 0; r < 8; ++r) c[r] = bv;
#pragma unroll
    for (int kk = 0; kk < 16; ++kk) {
      const int k0 = kk * 4 + lh * 2;
      v2f a, bb;
      a[0] = Xs[(rt * 16 + ll) * XS_STRIDE + k0];
      a[1] = Xs[(rt * 16 + ll) * XS_STRIDE + k0 + 1];
      bb[0] = w_v[(k0 + 0) * 64 + col];
      bb[1] = w_v[(k0 + 1) * 64 + col];
      c = wmma4(a, bb, c);
    }
    float acc = 0.f;
#pragma unroll
    for (int r = 0; r < 8; ++r) {
      const int row = rt * 16 + r + 8 * lh;
      const float pr = Q3[row * 4 + 0] * wc0 + Q3[row * 4 + 1] * wc1 +
                       Q3[row * 4 + 2] * wc2 + bp;
      const float v = c[r] * Dw[row] + pr;
      acc += v * SMn[row * 8 + (col & 7)];
    }
    acc += __shfl_xor(acc, 16, 32);
    if (lh == 0) out[(size_t)(q0 + rt) * 64 + col] = acc;
  }
}

// =====================================================================
// Host launcher.
// Workspace layout (assumes ws_size >= ~29 MB):
//   [0,2MB)   idx  int[N*K]
//   [2,4MB)   dist float[N*K]
//   [4,12MB)  x_q  float[N*64]
//   [12,28MB) h    float[N*K*8]
//   [28MB..)  stat buffers (few KB)
// =====================================================================
extern "C" void kernel_launch(void* const* d_in, const int* in_sizes, int n_in,
                              void* d_out, int out_size, void* d_ws, size_t ws_size,
                              hipStream_t stream) {
  (void)in_sizes; (void)n_in; (void)out_size; (void)ws_size;
  const float* p      = (const float*)d_in[0];
  const float* x      = (const float*)d_in[2];
  const float* w_q    = (const float*)d_in[3];
  const float* b_q    = (const float*)d_in[4];
  const float* w_k    = (const float*)d_in[5];
  const float* b_k    = (const float*)d_in[6];
  const float* w_v    = (const float*)d_in[7];
  const float* b_v    = (const float*)d_in[8];
  const float* w_p0   = (const float*)d_in[9];
  const float* b_p0   = (const float*)d_in[10];
  const float* bnPg   = (const float*)d_in[11];
  const float* bnPb   = (const float*)d_in[12];
  const float* w_p1   = (const float*)d_in[13];
  const float* b_p1   = (const float*)d_in[14];
  const float* bnW0g  = (const float*)d_in[15];
  const float* bnW0b  = (const float*)d_in[16];
  const float* w_w0   = (const float*)d_in[17];
  const float* b_w0   = (const float*)d_in[18];
  const float* bnW1g  = (const float*)d_in[19];
  const float* bnW1b  = (const float*)d_in[20];
  const float* w_w1   = (const float*)d_in[21];
  const float* b_w1   = (const float*)d_in[22];
  float* out = (float*)d_out;

  char* ws = (char*)d_ws;
  int*   IDX  = (int*)(ws);
  float* DIST = (float*)(ws + (size_t)2  * 1024 * 1024);
  float* XQ   = (float*)(ws + (size_t)4  * 1024 * 1024);
  float* HB   = (float*)(ws + (size_t)12 * 1024 * 1024);
  float* ST   = (float*)(ws + (size_t)28 * 1024 * 1024);
  float* sumP  = ST + 0,   *sqP  = ST + 3;
  float* sumW0 = ST + 8,   *sqW0 = ST + 72;
  float* sumW1 = ST + 136, *sqW1 = ST + 144;
  float* scP   = ST + 160, *shP  = ST + 164;
  float* scW0  = ST + 168, *shW0 = ST + 232;
  float* scW1  = ST + 296, *shW1 = ST + 304;

  const int LDS_W0  = (256 * XS_STRIDE + 256 * 4 + 128) * 4;
  const int LDS_W1  = (256 * XS_STRIDE + 256 * 4 + 256 * XS_STRIDE + 16) * 4;
  const int LDS_FIN = (256 * XS_STRIDE + 256 * 4 + 256 + 256 * 8 + 256 * 8) * 4;
  (void)hipFuncSetAttribute((const void*)stats_w0_kernel,
                            hipFuncAttributeMaxDynamicSharedMemorySize, LDS_W0);
  (void)hipFuncSetAttribute((const void*)stats_w1_kernel,
                            hipFuncAttributeMaxDynamicSharedMemorySize, LDS_W1);
  (void)hipFuncSetAttribute((const void*)final_kernel,
                            hipFuncAttributeMaxDynamicSharedMemorySize, LDS_FIN);

  const float invCount = 1.0f / (float)ROWS;

  hipMemsetAsync(ST, 0, 152 * sizeof(float), stream);          // stat accumulators
  knn_kernel<<<NPTS / 256, 256, 0, stream>>>(p, IDX, DIST);
  xq_gemm_kernel<<<(NPTS / 16) * 4 / 4, 128, 0, stream>>>(x, w_q, b_q, XQ);
  stats_p_kernel<<<512, 256, 0, stream>>>(p, IDX, w_p0, b_p0, ST);
  finalize_bn_kernel<<<1, 64, 0, stream>>>(sumP, sqP, bnPg, bnPb, scP, shP, 3, invCount);
  stats_w0_kernel<<<NPTS / 16, 256, LDS_W0, stream>>>(
      p, x, IDX, XQ, w_p0, b_p0, w_p1, b_p1, w_k, b_k, scP, shP, sumW0);
  finalize_bn_kernel<<<1, 64, 0, stream>>>(sumW0, sqW0, bnW0g, bnW0b, scW0, shW0, 64, invCount);
  stats_w1_kernel<<<NPTS / 16, 256, LDS_W1, stream>>>(
      p, x, IDX, XQ, w_p0, b_p0, w_p1, b_p1, w_k, b_k, scP, shP, scW0, shW0,
      w_w0, b_w0, HB, sumW1);
  finalize_bn_kernel<<<1, 64, 0, stream>>>(sumW1, sqW1, bnW1g, bnW1b, scW1, shW1, 8, invCount);
  final_kernel<<<NPTS / 16, 256, LDS_FIN, stream>>>(
      p, x, IDX, DIST, HB, w_p0, b_p0, w_p1, b_p1, w_v, b_v, w_w1, b_w1,
      scP, shP, scW1, shW1, out);
}